// TKANEncoder_38414187495783
// MI455X (gfx1250) — compile-verified
//
#include <hip/hip_runtime.h>

// TKAN encoder for MI455X (gfx1250, wave32).
// 4 persistent workgroups, one per 16-row batch tile. All state in LDS.
// Every GEMM uses V_WMMA_F32_16X16X4_F32 (exact fp32 on the matrix pipe).
// x_t tiles are double-buffered in LDS via GLOBAL_LOAD_ASYNC_TO_LDS_B128.

typedef __attribute__((ext_vector_type(2))) float v2f;
typedef __attribute__((ext_vector_type(8))) float v8f;

#define Tt  512
#define Ff  128
#define Hh  512
#define NBb 11
#define FNB (Ff * NBb)   // 1408

// Padded LDS row strides (floats) to avoid 64-bank conflicts on column walks.
#define XT_S 132
#define AG_S 132
#define H_S  516
#define BA_S 1412

// LDS layout (float offsets)
#define OFF_XT0  0
#define OFF_XT1  (OFF_XT0 + 16 * XT_S)
#define OFF_AGG  (OFF_XT1 + 16 * XT_S)
#define OFF_H0   (OFF_AGG + 16 * AG_S)
#define OFF_H1   (OFF_H0  + 16 * H_S)
#define OFF_S0   (OFF_H1  + 16 * H_S)
#define OFF_S1   (OFF_S0  + 16 * H_S)
#define OFF_KAN  (OFF_S1  + 16 * H_S)
#define OFF_BAS  (OFF_KAN + 16 * H_S)
#define SMEM_TOT (OFF_BAS + 16 * BA_S)   // 70208 floats = 274.25 KB (< 320 KB WGP LDS)

__device__ __forceinline__ float sigmoidf_(float v) {
    return 1.0f / (1.0f + __expf(-v));
}

// Async 16-byte copy global -> LDS, tracked by ASYNCcnt (CDNA5 async path).
__device__ __forceinline__ void async_g2l_b128(const float* lds_ptr,
                                               const float* __restrict__ gptr) {
    // LDS aperture keeps the wave-relative byte offset in addr[31:0].
    unsigned int lds_off = (unsigned int)(unsigned long long)(const void*)lds_ptr;
    asm volatile("global_load_async_to_lds_b128 %0, %1, off"
                 :
                 : "v"(lds_off), "v"(gptr)
                 : "memory");
}

__device__ __forceinline__ void wait_asynccnt0() {
    asm volatile("s_wait_asynccnt 0x0" ::: "memory");
}

// --- WMMA helpers: A (16xK) in LDS, B (KxN) in global, fp32 accumulators ---

// One output tile.
__device__ __forceinline__ void wmma_k1(const float* aLds, int aStride,
                                        const float* __restrict__ Bg, int ldb,
                                        int col0, int Ktot, v8f& acc0, int lane) {
    const int mn = lane & 15;
    const int kq = (lane >> 4) << 1;
    for (int k = 0; k < Ktot; k += 4) {
        v2f a, b0;
        a.x = aLds[mn * aStride + k + kq + 0];
        a.y = aLds[mn * aStride + k + kq + 1];
        const float* br = Bg + (size_t)(k + kq) * ldb + mn;
        b0.x = br[col0];
        b0.y = br[col0 + ldb];
        acc0 = __builtin_amdgcn_wmma_f32_16x16x4_f32(false, a, false, b0,
                                                     (short)0, acc0, false, false);
    }
}

// Two column tiles sharing one A fragment.
__device__ __forceinline__ void wmma_k2(const float* aLds, int aStride,
                                        const float* __restrict__ Bg, int ldb,
                                        int col0, int col1, int Ktot,
                                        v8f& acc0, v8f& acc1, int lane) {
    const int mn = lane & 15;
    const int kq = (lane >> 4) << 1;
    for (int k = 0; k < Ktot; k += 4) {
        v2f a, b0, b1;
        a.x = aLds[mn * aStride + k + kq + 0];
        a.y = aLds[mn * aStride + k + kq + 1];
        const float* br = Bg + (size_t)(k + kq) * ldb + mn;
        b0.x = br[col0];
        b0.y = br[col0 + ldb];
        b1.x = br[col1];
        b1.y = br[col1 + ldb];
        acc0 = __builtin_amdgcn_wmma_f32_16x16x4_f32(false, a, false, b0,
                                                     (short)0, acc0, false, false);
        acc1 = __builtin_amdgcn_wmma_f32_16x16x4_f32(false, a, false, b1,
                                                     (short)0, acc1, false, false);
    }
}

// Three weight matrices x two column tiles, all sharing one A fragment per k-step.
// 6 independent WMMA chains -> deep latency hiding; 3x fewer LDS A reads.
__device__ __forceinline__ void wmma_k6(const float* aLds, int aStride,
                                        const float* __restrict__ B0g,
                                        const float* __restrict__ B1g,
                                        const float* __restrict__ B2g,
                                        int ldb, int col0, int col1, int Ktot,
                                        v8f& p0, v8f& p1, v8f& q0, v8f& q1,
                                        v8f& r0, v8f& r1, int lane) {
    const int mn = lane & 15;
    const int kq = (lane >> 4) << 1;
    for (int k = 0; k < Ktot; k += 4) {
        v2f a;
        a.x = aLds[mn * aStride + k + kq + 0];
        a.y = aLds[mn * aStride + k + kq + 1];
        const size_t ro = (size_t)(k + kq) * ldb + mn;
        v2f b;
        b.x = B0g[ro + col0]; b.y = B0g[ro + col0 + ldb];
        p0 = __builtin_amdgcn_wmma_f32_16x16x4_f32(false, a, false, b, (short)0, p0, false, false);
        b.x = B0g[ro + col1]; b.y = B0g[ro + col1 + ldb];
        p1 = __builtin_amdgcn_wmma_f32_16x16x4_f32(false, a, false, b, (short)0, p1, false, false);
        b.x = B1g[ro + col0]; b.y = B1g[ro + col0 + ldb];
        q0 = __builtin_amdgcn_wmma_f32_16x16x4_f32(false, a, false, b, (short)0, q0, false, false);
        b.x = B1g[ro + col1]; b.y = B1g[ro + col1 + ldb];
        q1 = __builtin_amdgcn_wmma_f32_16x16x4_f32(false, a, false, b, (short)0, q1, false, false);
        b.x = B2g[ro + col0]; b.y = B2g[ro + col0 + ldb];
        r0 = __builtin_amdgcn_wmma_f32_16x16x4_f32(false, a, false, b, (short)0, r0, false, false);
        b.x = B2g[ro + col1]; b.y = B2g[ro + col1 + ldb];
        r1 = __builtin_amdgcn_wmma_f32_16x16x4_f32(false, a, false, b, (short)0, r1, false, false);
    }
}

// Spill one C/D-layout tile to LDS (C/D: lane n = lane&15, rows r + 8*(lane>>4)).
__device__ __forceinline__ void store_tile(float* dst, int stride, int col,
                                           const v8f& acc, int lane) {
    const int n  = lane & 15;
    const int mh = (lane >> 4) * 8;
#pragma unroll
    for (int r = 0; r < 8; ++r)
        dst[(mh + r) * stride + col + n] = acc[r];
}

__global__ __launch_bounds__(512, 1)
void tkan_encoder_kernel(const float* __restrict__ x,
                         const float* __restrict__ Wi, const float* __restrict__ Ui,
                         const float* __restrict__ bi,
                         const float* __restrict__ Wf, const float* __restrict__ Uf,
                         const float* __restrict__ bf,
                         const float* __restrict__ Wo, const float* __restrict__ Uo,
                         const float* __restrict__ bo,
                         const float* __restrict__ Wsx, const float* __restrict__ Wsh,
                         const float* __restrict__ base_w,
                         const float* __restrict__ spline_w,
                         const float* __restrict__ Whh, const float* __restrict__ Whz,
                         float* __restrict__ out) {
    __shared__ float sm[SMEM_TOT];

    const int tid  = threadIdx.x;
    const int lane = tid & 31;
    const int wave = tid >> 5;
    const int b0   = blockIdx.x * 16;      // this WG's batch rows

    // Zero all LDS state.
    for (int i = tid; i < SMEM_TOT; i += 512) sm[i] = 0.0f;
    __syncthreads();

    // Per-wave fixed output column pair for all H-wide GEMMs.
    const int colA = wave * 32;
    const int colB = colA + 16;

    // Cell state c lives in registers for the two owned tiles.
    v8f c0 = {}, c1 = {};

    // x staging: one b128 transfer per thread covers the 16x128 tile.
    const int xm = tid >> 5;          // row 0..15
    const int xf = (tid & 31) * 4;    // col 0..124 step 4

    // Preload x_0 into buffer 0 (synchronous path, once).
    {
        const float* xb = x + (size_t)b0 * Tt * Ff;
        for (int e = tid; e < 16 * Ff; e += 512) {
            const int m = e >> 7, f = e & 127;
            sm[OFF_XT0 + m * XT_S + f] = xb[(size_t)m * Tt * Ff + f];
        }
    }
    __syncthreads();

    int cur = 0;
    for (int t = 0; t < Tt; ++t) {
        float* hCur  = sm + (cur ? OFF_H1  : OFF_H0);
        float* hNxt  = sm + (cur ? OFF_H0  : OFF_H1);
        float* sCur  = sm + (cur ? OFF_S1  : OFF_S0);
        float* sNxt  = sm + (cur ? OFF_S0  : OFF_S1);
        float* xtCur = sm + (cur ? OFF_XT1 : OFF_XT0);
        float* xtNxt = sm + (cur ? OFF_XT0 : OFF_XT1);

        // ---- stage A: kick off async staging of x_{t+1} (overlaps whole step) ----
        if (t + 1 < Tt) {
            const float* xg = x + ((size_t)(b0 + xm) * Tt + (t + 1)) * Ff + xf;
            async_g2l_b128(xtNxt + xm * XT_S + xf, xg);
        }

        // ---- stage B: agg = x_t @ Wsx + sub_h @ Wsh  (16 x 128, 8 tiles) ----
        if (wave < 8) {
            v8f ag = {};
            const int col = wave * 16;
            wmma_k1(xtCur, XT_S, Wsx, Ff, col, Ff, ag, lane);
            wmma_k1(sCur,  H_S,  Wsh, Ff, col, Hh, ag, lane);
            store_tile(sm + OFF_AGG, AG_S, col, ag, lane);
        }
        __syncthreads();

        // ---- stage C: Cox-de Boor basis (K=3, 15 uniform knots) + SiLU ----
        for (int e = tid; e < 16 * Ff; e += 512) {
            const int m = e >> 7, f = e & 127;
            const float xv = sm[OFF_AGG + m * AG_S + f];
            float bb[14];
#pragma unroll
            for (int j = 0; j < 14; ++j) {
                const float t0 = 0.25f * j - 1.75f;
                bb[j] = (xv >= t0 && xv < t0 + 0.25f) ? 1.0f : 0.0f;
            }
            // p = 1 (1/h = 4)
#pragma unroll
            for (int j = 0; j < 13; ++j) {
                const float tj = 0.25f * j - 1.75f;
                bb[j] = (xv - tj) * 4.0f * bb[j] + (tj + 0.5f - xv) * 4.0f * bb[j + 1];
            }
            // p = 2 (1/(2h) = 2)
#pragma unroll
            for (int j = 0; j < 12; ++j) {
                const float tj = 0.25f * j - 1.75f;
                bb[j] = (xv - tj) * 2.0f * bb[j] + (tj + 0.75f - xv) * 2.0f * bb[j + 1];
            }
            // p = 3 (1/(3h) = 4/3)
#pragma unroll
            for (int j = 0; j < 11; ++j) {
                const float tj = 0.25f * j - 1.75f;
                bb[j] = (xv - tj) * (4.0f / 3.0f) * bb[j] +
                        (tj + 1.0f - xv) * (4.0f / 3.0f) * bb[j + 1];
            }
#pragma unroll
            for (int j = 0; j < NBb; ++j)
                sm[OFF_BAS + m * BA_S + f * NBb + j] = bb[j];
            // SiLU overwrites agg in place (each thread owns its elements)
            sm[OFF_AGG + m * AG_S + f] = xv * sigmoidf_(xv);
        }
        __syncthreads();

        // ---- stage D: fused gate + KAN GEMMs for this wave's 2 tiles ----
        v8f gi0, gi1, gf0, gf1, go0, go1, kn0, kn1;
        {
            const int n = lane & 15;
            const float vi0 = bi[colA + n], vi1 = bi[colB + n];
            const float vf0 = bf[colA + n], vf1 = bf[colB + n];
            const float vo0 = bo[colA + n], vo1 = bo[colB + n];
#pragma unroll
            for (int r = 0; r < 8; ++r) {
                gi0[r] = vi0; gi1[r] = vi1;
                gf0[r] = vf0; gf1[r] = vf1;
                go0[r] = vo0; go1[r] = vo1;
                kn0[r] = 0.0f; kn1[r] = 0.0f;
            }
        }
        // i/f/o share A fragments: one pass over x_t, one pass over h.
        wmma_k6(xtCur, XT_S, Wi, Wf, Wo, Hh, colA, colB, Ff,
                gi0, gi1, gf0, gf1, go0, go1, lane);
        wmma_k6(hCur,  H_S,  Ui, Uf, Uo, Hh, colA, colB, Hh,
                gi0, gi1, gf0, gf1, go0, go1, lane);
        // KAN candidate: silu(agg) @ base_w + basis @ spline_w.
        wmma_k2(sm + OFF_AGG, AG_S, base_w,   Hh, colA, colB, Ff,  kn0, kn1, lane);
        wmma_k2(sm + OFF_BAS, BA_S, spline_w, Hh, colA, colB, FNB, kn0, kn1, lane);

        // ---- stage E: elementwise LSTM-style cell update (register resident) ----
        {
            const int n  = lane & 15;
            const int mh = (lane >> 4) * 8;
            float* kanL = sm + OFF_KAN;
#pragma unroll
            for (int r = 0; r < 8; ++r) {
                // tile A
                {
                    const float iv = sigmoidf_(gi0[r]);
                    const float fv = sigmoidf_(gf0[r]);
                    const float ov = sigmoidf_(go0[r]);
                    const float cn = fv * c0[r] + iv * tanhf(kn0[r]);
                    c0[r] = cn;
                    hNxt[(mh + r) * H_S + colA + n] = ov * tanhf(cn);
                    kanL[(mh + r) * H_S + colA + n] = kn0[r];
                }
                // tile B
                {
                    const float iv = sigmoidf_(gi1[r]);
                    const float fv = sigmoidf_(gf1[r]);
                    const float ov = sigmoidf_(go1[r]);
                    const float cn = fv * c1[r] + iv * tanhf(kn1[r]);
                    c1[r] = cn;
                    hNxt[(mh + r) * H_S + colB + n] = ov * tanhf(cn);
                    kanL[(mh + r) * H_S + colB + n] = kn1[r];
                }
            }
        }
        __syncthreads();

        // ---- stage F: sub_h' = sub_h @ Whh + kan @ Whz ----
        {
            v8f s0 = {}, s1 = {};
            wmma_k2(sCur,         H_S, Whh, Hh, colA, colB, Hh, s0, s1, lane);
            wmma_k2(sm + OFF_KAN, H_S, Whz, Hh, colA, colB, Hh, s0, s1, lane);
            store_tile(sNxt, H_S, colA, s0, lane);
            store_tile(sNxt, H_S, colB, s1, lane);
        }

        // Async x_{t+1} staging must be complete (and made visible by the
        // barrier below) before the next step's GEMMs read it.
        wait_asynccnt0();
        __syncthreads();

        cur ^= 1;
    }

    // ---- epilogue: final h -> d_out [64, 512] ----
    const float* hFin = sm + (cur ? OFF_H1 : OFF_H0);
    for (int e = tid; e < 16 * Hh; e += 512) {
        const int m = e >> 9, n = e & 511;
        out[(size_t)(b0 + m) * Hh + n] = hFin[m * H_S + n];
    }
}

extern "C" void kernel_launch(void* const* d_in, const int* in_sizes, int n_in,
                              void* d_out, int out_size, void* d_ws, size_t ws_size,
                              hipStream_t stream) {
    const float* x        = (const float*)d_in[0];
    const float* Wi       = (const float*)d_in[1];
    const float* Ui       = (const float*)d_in[2];
    const float* bi       = (const float*)d_in[3];
    const float* Wf       = (const float*)d_in[4];
    const float* Uf       = (const float*)d_in[5];
    const float* bf       = (const float*)d_in[6];
    const float* Wo       = (const float*)d_in[7];
    const float* Uo       = (const float*)d_in[8];
    const float* bo       = (const float*)d_in[9];
    const float* Wsx      = (const float*)d_in[10];
    const float* Wsh      = (const float*)d_in[11];
    const float* base_w   = (const float*)d_in[12];
    const float* spline_w = (const float*)d_in[13];
    const float* Whh      = (const float*)d_in[14];
    const float* Whz      = (const float*)d_in[15];
    float* out = (float*)d_out;

    // 4 batch tiles of 16 rows, 512 threads (16 wave32) each.
    tkan_encoder_kernel<<<4, 512, 0, stream>>>(
        x, Wi, Ui, bi, Wf, Uf, bf, Wo, Uo, bo,
        Wsx, Wsh, base_w, spline_w, Whh, Whz, out);
}